// Butterfly_1580547970089
// MI455X (gfx1250) — compile-verified
//
#include <hip/hip_runtime.h>

// Butterfly layer (batch=16384, n=1024, log_n=10, 1 stack, 1 block) for MI455X.
//
// The 10 butterfly stages factor into 3 rounds of 64 dense 16x16 matrices
// (stages 0-3: contiguous 16-blocks; stages 4-7: stride-16 sets; stages 8-9:
// 4x 4x4 blocks packed block-diagonally). build_mats_kernel composes them
// (192 KB in d_ws, L2-resident, shared by all tiles). butterfly_wmma_kernel
// applies them to 16-row batch tiles with chained V_WMMA_F32_16X16X4_F32,
// keeping the whole pipeline in LDS. HBM traffic = 64MB in + 64MB out
// -> ~5.5us roofline at 23.3 TB/s; 1.6 GFLOP on the matrix units is free.
//
// LDS layout: buf[row][pos], row pitch 1028 floats (pad 4). pitch mod 64 == 4
// => B-fragment reads (fixed pos, 16 rows across lanes) hit banks 4*ln+pos:
// conflict-free, while keeping every LDS index AFFINE so all j/st offsets
// fold into DS immediate offsets (no per-access VALU address math).

typedef float v2f __attribute__((ext_vector_type(2)));
typedef float v8f __attribute__((ext_vector_type(8)));

#define PITCH 1028   // floats per LDS row

// ---------------------------------------------------------------------------
// Precompute: 192 blocks, each builds one 16x16 matrix (product of its
// butterfly stage factors) -> mats[(g*64+mi)*256 + r*16 + c].
// ---------------------------------------------------------------------------
__global__ __launch_bounds__(256) void build_mats_kernel(
        const float* __restrict__ tw, float* __restrict__ mats) {
    __shared__ float M[2][256];
    const int g  = blockIdx.x >> 6;   // 0: stages 0-3, 1: 4-7, 2: 8-9
    const int mi = blockIdx.x & 63;
    const int t  = threadIdx.x;

    M[0][t] = ((t >> 4) == (t & 15)) ? 1.0f : 0.0f;   // identity
    __syncthreads();

    int cur = 0;
    const int nst = (g == 2) ? 2 : 4;
    for (int s = 0; s < nst; ++s) {
        if (t < 128) {
            const int pp = t >> 4;    // butterfly pair 0..7
            const int cc = t & 15;    // column 0..15
            int m0, m1, bi, idx;
            if (g == 0) {
                idx = s;
                const int str = 1 << s;
                m0 = (pp / str) * 2 * str + (pp % str); m1 = m0 + str;
                bi = 8 * mi + pp;
            } else if (g == 1) {
                idx = 4 + s;
                const int str = 1 << s;
                m0 = (pp / str) * 2 * str + (pp % str); m1 = m0 + str;
                bi = 128 * (mi >> 4) + 16 * pp + (mi & 15);
            } else {
                idx = 8 + s;
                const int i = pp >> 1, h = pp & 1;
                if (s == 0) { m0 = i * 4 + h * 2; m1 = m0 + 1; }  // stride-256
                else        { m0 = i * 4 + h;     m1 = m0 + 2; }  // stride-512
                bi = 256 * h + 4 * mi + i;
            }
            const float* tq = tw + ((size_t)idx * 512 + bi) * 4;  // (10,512,2,2)
            const float t00 = tq[0], t01 = tq[1], t10 = tq[2], t11 = tq[3];
            const float x0 = M[cur][m0 * 16 + cc];
            const float x1 = M[cur][m1 * 16 + cc];
            M[cur ^ 1][m0 * 16 + cc] = t00 * x0 + t01 * x1;
            M[cur ^ 1][m1 * 16 + cc] = t10 * x0 + t11 * x1;
        }
        __syncthreads();
        cur ^= 1;
    }
    mats[(size_t)(g * 64 + mi) * 256 + t] = M[cur][t];
}

// ---------------------------------------------------------------------------
// One round: this wave executes 8 of the 64 16x16x16 matmuls (mi = w + 8j),
// in place in LDS. All index expressions are affine with compile-time j/st
// parts so DS/global offsets become immediates off two base registers.
// Positions: R0: 16mi+v   R1: 256(mi>>4)+16v+(mi&15)   R2: 4mi+(v>>2)+256(v&3)
// ---------------------------------------------------------------------------
template <int R>
__device__ __forceinline__ void do_round(float* __restrict__ buf,
                                         const float* __restrict__ mats,
                                         int w, int lane) {
    const int hi = lane >> 4;                     // K/M half select
    const int ln = lane & 15;
    float* bl = buf + ln * PITCH;                 // B/D row base for this lane
    const float* Mb = mats + (size_t)(R * 64 + w) * 256 + ln * 16;

#pragma unroll
    for (int j = 0; j < 8; ++j) {
        v2f a[4], b[4];
#pragma unroll
        for (int st = 0; st < 4; ++st) {
            const int kb = (st << 2) + (hi << 1);            // K base
            a[st] = *(const v2f*)(Mb + j * 2048 + kb);       // b64, imm offset
            if constexpr (R == 0) {
                b[st] = *(const v2f*)(bl + (w << 4) + kb + j * 128);
            } else if constexpr (R == 1) {
                const int base = w + 8 * (j & 1) + 256 * (j >> 1);
                b[st].x = bl[base + 16 * kb];
                b[st].y = bl[base + 16 * (kb + 1)];
            } else {
                const int base = 4 * w + 32 * j;
                b[st].x = bl[base + (kb >> 2) + ((kb & 3) << 8)];
                b[st].y = bl[base + ((kb + 1) >> 2) + (((kb + 1) & 3) << 8)];
            }
        }
        v8f acc = {0.f, 0.f, 0.f, 0.f, 0.f, 0.f, 0.f, 0.f};
#pragma unroll
        for (int st = 0; st < 4; ++st)
            acc = __builtin_amdgcn_wmma_f32_16x16x4_f32(
                false, a[st], false, b[st], (short)0, acc, false, false);
        // D: vgpr i holds M = i + 8*hi, N = ln. Same-wave DS ops are in
        // order and each matmul's 16-position set is private to it.
        if constexpr (R == 0) {
            float* d = bl + (w << 4) + (hi << 3) + j * 128;  // 8 consecutive pos
            float4 lo, hi4;
            lo.x = acc[0]; lo.y = acc[1]; lo.z = acc[2]; lo.w = acc[3];
            hi4.x = acc[4]; hi4.y = acc[5]; hi4.z = acc[6]; hi4.w = acc[7];
            *(float4*)(d)     = lo;                           // ds_store_b128
            *(float4*)(d + 4) = hi4;
        } else if constexpr (R == 1) {
            const int base = w + 8 * (j & 1) + 256 * (j >> 1) + 128 * hi;
#pragma unroll
            for (int i = 0; i < 8; ++i) bl[base + 16 * i] = acc[i];
        } else {
            const int base = 4 * w + 32 * j;
#pragma unroll
            for (int i = 0; i < 8; ++i) {
                const int m = i + (hi << 3);
                bl[base + (m >> 2) + ((m & 3) << 8)] = acc[i];
            }
        }
    }
}

// ---------------------------------------------------------------------------
// Main kernel: one workgroup (256 thr = 8 wave32) per 16-row batch tile.
// ---------------------------------------------------------------------------
__global__ __launch_bounds__(256) void butterfly_wmma_kernel(
        const float* __restrict__ in, const float* __restrict__ bias,
        const float* __restrict__ mats, float* __restrict__ out) {
    __shared__ float buf[16 * PITCH];  // 65792 B; ~4 workgroups / 320KB WGP
    const int t    = threadIdx.x;
    const int lane = t & 31;
    const int w    = t >> 5;
    const size_t tile_base = (size_t)blockIdx.x * 16 * 1024;
    const float* inT  = in  + tile_base;
    float*       outT = out + tile_base;

    // Phase 0: async-copy 16x1024 f32 tile to LDS; 16B per lane, coalesced,
    // contiguous on both sides (pad lives at end of each LDS row).
#pragma unroll
    for (int it = 0; it < 16; ++it) {
        const int idx = t + it * 256;          // 0..4095 b128 chunks
        const int row = idx >> 8;
        const int col = (idx & 255) << 2;
        const unsigned ldsb = (unsigned)((row * PITCH + col) * 4);
        const unsigned long long ga =
            (unsigned long long)(const void*)(inT + row * 1024 + col);
        asm volatile("global_load_async_to_lds_b128 %0, %1, off"
                     :: "v"(ldsb), "v"(ga) : "memory");
    }
    asm volatile("s_wait_asynccnt 0x0" ::: "memory");
    __syncthreads();

    do_round<0>(buf, mats, w, lane);
    __syncthreads();
    do_round<1>(buf, mats, w, lane);
    __syncthreads();
    do_round<2>(buf, mats, w, lane);
    __syncthreads();

    // Phase 3: bias + coalesced b128 stores.
#pragma unroll
    for (int it = 0; it < 16; ++it) {
        const int idx = t + it * 256;
        const int row = idx >> 8;
        const int col = (idx & 255) << 2;
        const float4 vb = *(const float4*)(bias + col);
        const float4 vx = *(const float4*)(&buf[row * PITCH + col]);
        float4 vo;
        vo.x = vx.x + vb.x; vo.y = vx.y + vb.y;
        vo.z = vx.z + vb.z; vo.w = vx.w + vb.w;
        *(float4*)(outT + row * 1024 + col) = vo;
    }
}

extern "C" void kernel_launch(void* const* d_in, const int* in_sizes, int n_in,
                              void* d_out, int out_size, void* d_ws, size_t ws_size,
                              hipStream_t stream) {
    const float* input   = (const float*)d_in[0];   // (16384, 1024) f32
    const float* twiddle = (const float*)d_in[1];   // (1,1,10,512,2,2) f32
    const float* bias    = (const float*)d_in[2];   // (1024,) f32
    float* o    = (float*)d_out;                    // (16384, 1024) f32
    float* mats = (float*)d_ws;                     // needs 192 KB scratch

    const int batch = in_sizes[0] / 1024;           // 16384
    build_mats_kernel<<<192, 256, 0, stream>>>(twiddle, mats);
    butterfly_wmma_kernel<<<batch / 16, 256, 0, stream>>>(input, bias, mats, o);
}